// MultiHeadSelfAttention_56736517980685
// MI455X (gfx1250) — compile-verified
//
#include <hip/hip_runtime.h>

// MI455X / gfx1250, wave32. fp32-exact path via V_WMMA_F32_16X16X4_F32.
// Attention has no softmax -> (Q K^T) V == Q (K^T V); K^T V is 64x64/head,
// so the 537MB attn intermediate and 33.3 of its 34.4 GF vanish.
// Projections (34.4 GF) run on a 128x128-tile double-buffered GEMM fed by
// GLOBAL_LOAD_ASYNC_TO_LDS_B128 (ASYNCcnt) ping-pong staging.

typedef float v2f __attribute__((ext_vector_type(2)));
typedef float v8f __attribute__((ext_vector_type(8)));

#define WMMA_F32(a, b, c) \
    __builtin_amdgcn_wmma_f32_16x16x4_f32(false, (a), false, (b), (short)0, (c), false, false)

// V_WMMA_F32_16X16X4_F32 fragment layouts (wave32):
//  A (16x4): VGPR v, lane l -> A[m = l&15][k = 2*(l>>4) + v]
//  B (4x16): VGPR v, lane l -> B[k = 2*(l>>4) + v][n = l&15]
//  C/D     : VGPR v, lane l -> C[m = v + 8*(l>>4)][n = l&15]

static constexpr int BATCH = 2;
static constexpr int SEQ   = 2048;
static constexpr int HID   = 1024;
static constexpr int NHEAD = 16;
static constexpr int HD    = 64;

// Async copy global->LDS, 16 bytes per lane, GVS addressing (saddr + vaddr32).
__device__ __forceinline__ void async_b128_to_lds(unsigned lds_byte,
                                                  const float* base,
                                                  unsigned byte_off) {
  asm volatile("global_load_async_to_lds_b128 %0, %1, %2"
               :: "v"(lds_byte), "v"(byte_off), "s"(base) : "memory");
}
__device__ __forceinline__ void wait_asynccnt_le(int n) {
  if (n == 0) asm volatile("s_wait_asynccnt 0" ::: "memory");
  else        asm volatile("s_wait_asynccnt 4" ::: "memory");
}

// ---------------------------------------------------------------------------
// C(M x N) = A(M x K, lda) * Bg(N x K, ldb)^T, fp32.
// Block tile 128x128, 256 threads = 8 waves (4x2); wave owns 32 rows x 64 cols
// = 2x4 accumulators. K staged 16-wide, double-buffered via async-to-LDS.
// Row stride 20 floats: 16B-aligned rows (async b128) + conflict-free b64 reads.
// ---------------------------------------------------------------------------
__global__ __launch_bounds__(256) void gemm_bt128_kernel(
    const float* __restrict__ A, int lda,
    const float* __restrict__ Bg, int ldb,
    float* __restrict__ C, int ldc, int K) {
  constexpr int BM = 128, BN = 128, KC = 16, LSTR = 20;

  const int tid  = threadIdx.x;
  const int wave = tid >> 5;
  const int lane = tid & 31;
  const int hl   = lane >> 4;
  const int r    = lane & 15;
  const int wm   = wave & 3;   // 4 row groups of 32
  const int wn   = wave >> 2;  // 2 col groups of 64

  const int m0 = blockIdx.x * BM;
  const int n0 = blockIdx.y * BN;

  __shared__ float As[2][BM * LSTR];  // 2 x 10KB
  __shared__ float Bs[2][BN * LSTR];  // 2 x 10KB

  // Per-thread staging slots: 512 float4 per tile, 2 each for A and B.
  const int srow0 = tid >> 2;            // i = tid      -> rows 0..63
  const int srow1 = (tid + 256) >> 2;    // i = tid+256  -> rows 64..127
  const int sc4   = (tid & 3) * 4;

  auto stage = [&](int buf, int k0) {
    const unsigned aL = (unsigned)(unsigned long long)&As[buf][0];
    const unsigned bL = (unsigned)(unsigned long long)&Bs[buf][0];
    async_b128_to_lds(aL + (srow0 * LSTR + sc4) * 4u, A,
                      (unsigned)((m0 + srow0) * lda + k0 + sc4) * 4u);
    async_b128_to_lds(aL + (srow1 * LSTR + sc4) * 4u, A,
                      (unsigned)((m0 + srow1) * lda + k0 + sc4) * 4u);
    async_b128_to_lds(bL + (srow0 * LSTR + sc4) * 4u, Bg,
                      (unsigned)((n0 + srow0) * ldb + k0 + sc4) * 4u);
    async_b128_to_lds(bL + (srow1 * LSTR + sc4) * 4u, Bg,
                      (unsigned)((n0 + srow1) * ldb + k0 + sc4) * 4u);
  };

  v8f acc[2][4] = {};

  stage(0, 0);
  int buf = 0;
  for (int k0 = 0; k0 < K; k0 += KC, buf ^= 1) {
    const int nxt = k0 + KC;
    __syncthreads();               // prev readers of buf^1 are done
    if (nxt < K) stage(buf ^ 1, nxt);
    wait_asynccnt_le(nxt < K ? 4 : 0);  // newest 4 belong to next chunk (in-order)
    __syncthreads();               // all waves' chunk-k0 data visible

#pragma unroll
    for (int kk = 0; kk < KC; kk += 4) {
      const int kf = kk + 2 * hl;  // even -> aligned b64 pair
      v2f aF[2];
#pragma unroll
      for (int t = 0; t < 2; ++t) {
        const float* ap = &As[buf][(wm * 32 + t * 16 + r) * LSTR + kf];
        aF[t].x = ap[0]; aF[t].y = ap[1];
      }
#pragma unroll
      for (int u = 0; u < 4; ++u) {
        const float* bp = &Bs[buf][(wn * 64 + u * 16 + r) * LSTR + kf];
        v2f bF; bF.x = bp[0]; bF.y = bp[1];
        acc[0][u] = WMMA_F32(aF[0], bF, acc[0][u]);
        acc[1][u] = WMMA_F32(aF[1], bF, acc[1][u]);
      }
    }
  }

#pragma unroll
  for (int t = 0; t < 2; ++t)
#pragma unroll
    for (int u = 0; u < 4; ++u)
#pragma unroll
      for (int v = 0; v < 8; ++v)
        C[(long)(m0 + wm * 32 + t * 16 + v + 8 * hl) * ldc +
          n0 + wn * 64 + u * 16 + r] = acc[t][u][v];
}

// ---------------------------------------------------------------------------
// Per-head out = Q_h * P^T (P = V^T K, 64x64). Block 128x64, blockIdx.z = b*16+h.
// ---------------------------------------------------------------------------
__global__ __launch_bounds__(256) void attn_gemm_kernel(
    const float* __restrict__ A, int lda,
    const float* __restrict__ Bg, int ldb,
    float* __restrict__ C, int ldc, int K) {
  constexpr int BM = 128, BN = 64, KC = 16, LDSP = KC + 2;

  const int z = blockIdx.z, b = z >> 4, hh = z & 15;
  A  += (long)b * SEQ * lda + hh * HD;
  C  += (long)b * SEQ * ldc + hh * HD;
  Bg += (long)z * HD * HD;

  const int tid = threadIdx.x;
  const int wave = tid >> 5;
  const int lane = tid & 31;
  const int hl = lane >> 4, r = lane & 15;
  const int m0 = blockIdx.x * BM;

  __shared__ float As[BM][LDSP];
  __shared__ float Bs[BN][LDSP];

  v8f acc[BN / 16] = {};

  for (int k0 = 0; k0 < K; k0 += KC) {
    __syncthreads();
    for (int i = tid; i < BM * KC / 4; i += 256) {
      const int row = i >> 2, c4 = (i & 3) * 4;
      const float4 val = *(const float4*)(A + (long)(m0 + row) * lda + k0 + c4);
      As[row][c4] = val.x; As[row][c4 + 1] = val.y;
      As[row][c4 + 2] = val.z; As[row][c4 + 3] = val.w;
    }
    for (int i = tid; i < BN * KC / 4; i += 256) {
      const int row = i >> 2, c4 = (i & 3) * 4;
      const float4 val = *(const float4*)(Bg + (long)(row) * ldb + k0 + c4);
      Bs[row][c4] = val.x; Bs[row][c4 + 1] = val.y;
      Bs[row][c4 + 2] = val.z; Bs[row][c4 + 3] = val.w;
    }
    __syncthreads();
#pragma unroll
    for (int kk = 0; kk < KC; kk += 4) {
      const int kf = kk + 2 * hl;
      v2f aF;
      aF.x = As[wave * 16 + r][kf];
      aF.y = As[wave * 16 + r][kf + 1];
#pragma unroll
      for (int t = 0; t < BN / 16; ++t) {
        v2f bF;
        bF.x = Bs[t * 16 + r][kf];
        bF.y = Bs[t * 16 + r][kf + 1];
        acc[t] = WMMA_F32(aF, bF, acc[t]);
      }
    }
  }
#pragma unroll
  for (int t = 0; t < BN / 16; ++t)
#pragma unroll
    for (int v = 0; v < 8; ++v)
      C[(long)(m0 + wave * 16 + v + 8 * hl) * ldc + t * 16 + r] = acc[t][v];
}

// ---------------------------------------------------------------------------
// P[b,h] = V_h^T K_h (64x64); one block per (b,h), 16 tiles, 2 per wave.
// ---------------------------------------------------------------------------
__global__ __launch_bounds__(256) void kvT_kernel(
    const float* __restrict__ v, const float* __restrict__ k,
    float* __restrict__ P) {
  const int z = blockIdx.x, b = z >> 4, hh = z & 15;
  const int tid = threadIdx.x;
  const int wave = tid >> 5;
  const int lane = tid & 31;
  const int hl = lane >> 4, r = lane & 15;
  const int it = wave & 3;
  const int jp = wave >> 2;

  const float* __restrict__ Vh = v + (long)b * SEQ * HID + hh * HD;
  const float* __restrict__ Kh = k + (long)b * SEQ * HID + hh * HD;

  v8f acc0 = {}, acc1 = {};
#pragma unroll 4
  for (int m = 0; m < SEQ; m += 4) {
    const long row0 = m + 2 * hl;
    v2f aF, b0, b1;
    aF.x = Vh[row0 * HID + it * 16 + r];
    aF.y = Vh[(row0 + 1) * HID + it * 16 + r];
    b0.x = Kh[row0 * HID + (2 * jp + 0) * 16 + r];
    b0.y = Kh[(row0 + 1) * HID + (2 * jp + 0) * 16 + r];
    b1.x = Kh[row0 * HID + (2 * jp + 1) * 16 + r];
    b1.y = Kh[(row0 + 1) * HID + (2 * jp + 1) * 16 + r];
    acc0 = WMMA_F32(aF, b0, acc0);
    acc1 = WMMA_F32(aF, b1, acc1);
  }

  float* __restrict__ Pz = P + (long)z * HD * HD;
#pragma unroll
  for (int vv = 0; vv < 8; ++vv) {
    const int iRow = it * 16 + vv + 8 * hl;
    Pz[iRow * HD + (2 * jp + 0) * 16 + r] = acc0[vv];
    Pz[iRow * HD + (2 * jp + 1) * 16 + r] = acc1[vv];
  }
}

// ---------------------------------------------------------------------------
extern "C" void kernel_launch(void* const* d_in, const int* in_sizes, int n_in,
                              void* d_out, int out_size, void* d_ws, size_t ws_size,
                              hipStream_t stream) {
  const float* h  = (const float*)d_in[0];
  // d_in[1] = key_pe: dead branch in the reference
  const float* Wq = (const float*)d_in[2];
  const float* Wk = (const float*)d_in[3];
  const float* Wv = (const float*)d_in[4];
  const float* Wo = (const float*)d_in[5];
  float* out = (float*)d_out;

  float* ws = (float*)d_ws;
  const long NE = (long)BATCH * SEQ * HID;
  float* q  = ws;
  float* k  = ws + NE;
  float* v  = ws + 2 * NE;
  float* ao = ws + 3 * NE;
  float* P  = ws + 4 * NE;  // 32 * 64 * 64

  const dim3 blk(256);
  const dim3 gProj(BATCH * SEQ / 128, HID / 128);  // 32 x 8
  gemm_bt128_kernel<<<gProj, blk, 0, stream>>>(h, HID, Wq, HID, q, HID, HID);
  gemm_bt128_kernel<<<gProj, blk, 0, stream>>>(h, HID, Wk, HID, k, HID, HID);
  gemm_bt128_kernel<<<gProj, blk, 0, stream>>>(h, HID, Wv, HID, v, HID, HID);

  kvT_kernel<<<dim3(BATCH * NHEAD), blk, 0, stream>>>(v, k, P);

  const dim3 gAttn(SEQ / 128, 1, BATCH * NHEAD);
  attn_gemm_kernel<<<gAttn, blk, 0, stream>>>(q, HID, P, HD, ao, HID, HD);

  gemm_bt128_kernel<<<gProj, blk, 0, stream>>>(ao, HID, Wo, HID, out, HID, HID);
}